// AlignLoss_3839700763192
// MI455X (gfx1250) — compile-verified
//
#include <hip/hip_runtime.h>

#define BATCH 4
#define WSZ   512
#define HW    (WSZ * WSZ)
#define KSEG  256

typedef __attribute__((ext_vector_type(2))) float v2f;
typedef __attribute__((ext_vector_type(8))) float v8f;

// ---- workspace layout (floats) ----
#define OFF_STATS 0                         // B*K*8 segment sums
#define OFF_H     (BATCH * KSEG * 8)        // B*K*12 : 9 Hinv + valid + pad
#define OFF_LOSS  (OFF_H + BATCH * KSEG * 12)
#define WS_FLOATS (OFF_LOSS + 2)

// ------------------------------------------------------------------
// 0) zero the workspace accumulators (harness poisons d_ws)
// ------------------------------------------------------------------
__global__ void k_init(float* __restrict__ ws) {
    int i = blockIdx.x * blockDim.x + threadIdx.x;
    if (i < WS_FLOATS) ws[i] = 0.0f;
}

// ------------------------------------------------------------------
// 1) per-(batch,label) segment sums via LDS histograms
// ------------------------------------------------------------------
__global__ void __launch_bounds__(256)
k_stats(const int* __restrict__ labels, const float* __restrict__ inj,
        const float* __restrict__ trs, const float* __restrict__ rot,
        const float* __restrict__ sca, float* __restrict__ ws) {
    __shared__ float s[KSEG * 8];
    for (int i = threadIdx.x; i < KSEG * 8; i += 256) s[i] = 0.0f;
    __syncthreads();

    const int PPB = 8192;                       // pixels per block (divides HW)
    int base  = blockIdx.x * PPB;
    int b     = base / HW;
    int pbase = base - b * HW;
    for (int j = 0; j < PPB / 256; ++j) {
        int pix = pbase + j * 256 + threadIdx.x;
        int lab = labels[b * HW + pix];
        if (lab > 0) {
            int sg = lab - 1;
            float colf = (float)(pix & (WSZ - 1));
            float rowf = (float)(pix >> 9);
            float* e = &s[sg * 8];
            atomicAdd(&e[0], 1.0f);
            atomicAdd(&e[1], colf);
            atomicAdd(&e[2], rowf);
            atomicAdd(&e[3], inj[b * HW + pix]);
            atomicAdd(&e[4], trs[(b * 2 + 0) * HW + pix]);
            atomicAdd(&e[5], trs[(b * 2 + 1) * HW + pix]);
            atomicAdd(&e[6], rot[b * HW + pix]);
            atomicAdd(&e[7], sca[b * HW + pix]);
        }
    }
    __syncthreads();
    float* dst = ws + OFF_STATS + b * KSEG * 8;
    for (int i = threadIdx.x; i < KSEG * 8; i += 256)
        if (s[i] != 0.0f) atomicAdd(&dst[i], s[i]);
}

// ------------------------------------------------------------------
// 2) compose homography chain, invert, store Hinv + valid
// ------------------------------------------------------------------
__device__ inline void mm3(const float* A, const float* B, float* C) {
#pragma unroll
    for (int i = 0; i < 3; ++i)
#pragma unroll
        for (int j = 0; j < 3; ++j)
            C[i*3+j] = A[i*3+0]*B[0*3+j] + A[i*3+1]*B[1*3+j] + A[i*3+2]*B[2*3+j];
}

__global__ void k_homog(float* __restrict__ ws) {
    int idx = blockIdx.x * blockDim.x + threadIdx.x;
    if (idx >= BATCH * KSEG) return;
    const float* st = ws + OFF_STATS + idx * 8;
    float cnt = st[0];
    float inv = 1.0f / fmaxf(cnt, 1.0f);
    float bx = st[1]*inv, by = st[2]*inv, rem = st[3]*inv;
    float ti = st[4]*inv, tj = st[5]*inv, rr = st[6]*inv, sc = st[7]*inv;
    bool valid = (cnt > 0.0f) && (rem < 0.5f);
    const float half = (float)(WSZ / 2);
    float bxn = (half - bx) / half, byn = (half - by) / half;
    float c = cosf(rr), sn = sinf(rr);
    float R [9] = {c, -sn, 0,  sn, c, 0,  0, 0, 1};
    float T [9] = {1, 0, ti,   0, 1, tj,  0, 0, 1};
    float S [9] = {1.0f + sc, 0, 0,  0, 1.0f + sc, 0,  0, 0, 1};
    float Bp[9] = {1, 0, bxn,  0, 1, byn,   0, 0, 1};
    float Bq[9] = {1, 0, -bxn, 0, 1, -byn,  0, 0, 1};
    float M1[9], M2[9], M3[9], H[9];
    mm3(R, Bp, M1); mm3(S, M1, M2); mm3(Bq, M2, M3); mm3(T, M3, H);
    // inverse via adjugate
    float a = H[0], b2 = H[1], c2 = H[2], d = H[3], e = H[4], f = H[5],
          g = H[6], h = H[7], i9 = H[8];
    float A0 = e*i9 - f*h, A1 = -(d*i9 - f*g), A2 = d*h - e*g;
    float id = 1.0f / (a*A0 + b2*A1 + c2*A2);
    float* o = ws + OFF_H + idx * 12;
    o[0] = A0*id;  o[1] = -(b2*i9 - c2*h)*id;  o[2] =  (b2*f - c2*e)*id;
    o[3] = A1*id;  o[4] =  (a*i9  - c2*g)*id;  o[5] = -(a*f  - c2*d)*id;
    o[6] = A2*id;  o[7] = -(a*h   - b2*g)*id;  o[8] =  (a*e  - b2*d)*id;
    o[9] = valid ? 1.0f : 0.0f;  o[10] = 0.0f;  o[11] = 0.0f;
}

// ------------------------------------------------------------------
// 3) projection: 16 instances x 16 pixels per WMMA triple.
//    A(16x4): one Hinv row per instance; B(4x16): (gx,gy,1,0) per pixel.
//    v_wmma_f32_16x16x4_f32 gives den/u/v for 256 (inst,pixel) pairs.
// ------------------------------------------------------------------
__global__ void __launch_bounds__(256)
k_project(const int* __restrict__ labels, const float* __restrict__ ws,
          float* __restrict__ outp) {
    __shared__ float   sH[KSEG * 12];
    __shared__ unsigned smask[8];
    const int tilesPerBatch = HW / 128;   // 8 waves * 16 pixels per block
    int b    = blockIdx.x / tilesPerBatch;
    int tile = blockIdx.x % tilesPerBatch;

    const float* src = ws + OFF_H + b * KSEG * 12;
    for (int i = threadIdx.x; i < KSEG * 12; i += 256) sH[i] = src[i];
    __syncthreads();
    if (threadIdx.x < 8) {
        unsigned m = 0;
        for (int j = 0; j < 32; ++j)
            if (sH[(threadIdx.x * 32 + j) * 12 + 9] != 0.0f) m |= (1u << j);
        smask[threadIdx.x] = m;
    }
    __syncthreads();

    int lane   = threadIdx.x & 31;
    int wave   = threadIdx.x >> 5;
    int laneHi = (lane >> 4) & 1;        // lanes 16-31 carry K=2,3 / M=r+8
    int n      = lane & 15;              // pixel column of the WMMA tile
    int pix    = tile * 128 + wave * 16 + n;
    int col    = pix & (WSZ - 1);
    int row    = pix >> 9;
    float gx = -1.0f + 2.0f * (float)col * (1.0f / 511.0f);
    float gy = -1.0f + 2.0f * (float)row * (1.0f / 511.0f);
    // B(4x16): rows = {gx, gy, 1, 0}; VGPR0 holds K=0/K=2, VGPR1 holds K=1/K=3
    v2f bm;
    bm.x = laneHi ? 1.0f : gx;
    bm.y = laneHi ? 0.0f : gy;

    const int* labp = labels + b * HW;
    float acc = 0.0f;

    for (int chunk = 0; chunk < KSEG / 16; ++chunk) {
        const float* hp = &sH[(chunk * 16 + n) * 12];  // A row M = n
        v2f aU, aV, aD;
        aU.x = laneHi ? hp[2] : hp[0];  aU.y = laneHi ? 0.0f : hp[1];
        aV.x = laneHi ? hp[5] : hp[3];  aV.y = laneHi ? 0.0f : hp[4];
        aD.x = laneHi ? hp[8] : hp[6];  aD.y = laneHi ? 0.0f : hp[7];
        v8f cz = {0.f, 0.f, 0.f, 0.f, 0.f, 0.f, 0.f, 0.f};
        v8f dU = __builtin_amdgcn_wmma_f32_16x16x4_f32(false, aU, false, bm,
                                                       (short)0, cz, false, false);
        v8f dV = __builtin_amdgcn_wmma_f32_16x16x4_f32(false, aV, false, bm,
                                                       (short)0, cz, false, false);
        v8f dD = __builtin_amdgcn_wmma_f32_16x16x4_f32(false, aD, false, bm,
                                                       (short)0, cz, false, false);
        unsigned mw = smask[chunk >> 1] >> ((chunk & 1) * 16);
#pragma unroll
        for (int r = 0; r < 8; ++r) {          // D: VGPR r = instance r / r+8
            float rinv = __builtin_amdgcn_rcpf(dD[r]);
            float u = dU[r] * rinv;
            float v = dV[r] * rinv;
            int ix = __float2int_rn(fmaf(u, 255.5f, 255.5f));
            int iy = __float2int_rn(fmaf(v, 255.5f, 255.5f));
            bool inb = ((unsigned)ix < (unsigned)WSZ) && ((unsigned)iy < (unsigned)WSZ);
            int ixc = min(max(ix, 0), WSZ - 1);
            int iyc = min(max(iy, 0), WSZ - 1);
            int sl  = labp[(iyc << 9) + ixc];   // L2-resident gather (1MB/batch)
            int instId = chunk * 16 + r + laneHi * 8;
            bool vb = (mw >> (r + laneHi * 8)) & 1u;
            acc += (inb && vb && (sl == instId + 1)) ? 1.0f : 0.0f;
        }
    }
    acc += __shfl_xor(acc, 16, 32);             // merge instance halves
    if (lane < 16) outp[1 + b * HW + pix] = acc;
}

// ------------------------------------------------------------------
// 4) loss over crop + in-place proj -> boolean mask
// ------------------------------------------------------------------
__global__ void __launch_bounds__(256)
k_loss(const float* __restrict__ gti, float* __restrict__ outp,
       float* __restrict__ ws) {
    int gidx = blockIdx.x * 256 + threadIdx.x;
    float p = outp[1 + gidx];
    outp[1 + gidx] = (p != 0.0f) ? 1.0f : 0.0f;
    int b = gidx >> 18;
    int pix = gidx & (HW - 1);
    int row = pix >> 9, col = pix & (WSZ - 1);
    float se = 0.0f, ae = 0.0f;
    if (row >= 32 && row < WSZ - 32 && col >= 32 && col < WSZ - 32) {
        float d = p - gti[b * HW + pix];
        se = d * d;
        ae = fabsf(d);
    }
#pragma unroll
    for (int o = 16; o > 0; o >>= 1) {
        se += __shfl_down(se, o, 32);
        ae += __shfl_down(ae, o, 32);
    }
    __shared__ float sS[8], sA[8];
    if ((threadIdx.x & 31) == 0) { sS[threadIdx.x >> 5] = se; sA[threadIdx.x >> 5] = ae; }
    __syncthreads();
    if (threadIdx.x == 0) {
        float S = 0.0f, A = 0.0f;
#pragma unroll
        for (int i = 0; i < 8; ++i) { S += sS[i]; A += sA[i]; }
        atomicAdd(ws + OFF_LOSS + 0, S);
        atomicAdd(ws + OFF_LOSS + 1, A);
    }
}

__global__ void k_final(const float* __restrict__ ws, float* __restrict__ outp) {
    if (threadIdx.x == 0 && blockIdx.x == 0) {
        const float invN = 1.0f / (float)(BATCH * (WSZ - 64) * (WSZ - 64));
        outp[0] = (ws[OFF_LOSS] + ws[OFF_LOSS + 1]) * invN;
    }
}

// ------------------------------------------------------------------
extern "C" void kernel_launch(void* const* d_in, const int* in_sizes, int n_in,
                              void* d_out, int out_size, void* d_ws, size_t ws_size,
                              hipStream_t stream) {
    // setup_inputs order: rgb, mod, gti, seg_inj, trs, rot, sca, labels
    const float* gti    = (const float*)d_in[2];
    const float* inj    = (const float*)d_in[3];
    const float* trs    = (const float*)d_in[4];
    const float* rot    = (const float*)d_in[5];
    const float* sca    = (const float*)d_in[6];
    const int*   labels = (const int*)  d_in[7];
    float* out = (float*)d_out;
    float* ws  = (float*)d_ws;

    k_init   <<<(WS_FLOATS + 255) / 256,       256, 0, stream>>>(ws);
    k_stats  <<<(BATCH * HW) / 8192,           256, 0, stream>>>(labels, inj, trs, rot, sca, ws);
    k_homog  <<<(BATCH * KSEG + 255) / 256,    256, 0, stream>>>(ws);
    k_project<<<(BATCH * HW) / 128,            256, 0, stream>>>(labels, ws, out);
    k_loss   <<<(BATCH * HW) / 256,            256, 0, stream>>>(gti, out, ws);
    k_final  <<<1, 1, 0, stream>>>(ws, out);
}